// OhemCrossEntropy2d_7301444403369
// MI455X (gfx1250) — compile-verified
//
#include <hip/hip_runtime.h>
#include <hip/hip_bf16.h>

typedef __attribute__((ext_vector_type(2))) float v2f;
typedef __attribute__((ext_vector_type(8))) float v8f;

#define IGNORE_LBL 0
#define MIN_KEPT   200000u
#define THRESH     0.6f
#define NBATCH     8
#define CCH        19
#define HWSZ       (512 * 1024)   /* 524288 = 2^19 */
#define HW_SHIFT   19
#define NPIX       (NBATCH * HWSZ) /* 4194304 */

/* control block layout (uint32 indices inside ws) */
#define H1OFF 0
#define H2OFF 1024
#define H3OFF (1024 + 4096)
#define C_NV  (1024 + 4096 + 256)
#define C_K   (C_NV + 1)
#define C_B1  (C_NV + 2)
#define C_B2  (C_NV + 3)
#define C_TH  (C_NV + 4)
#define C_DO  (C_NV + 5)
#define C_SUM (C_NV + 6)
#define C_CNT (C_NV + 7)
#define C_TOTAL (C_NV + 8)

// lane <-> lane^16 exchange via ds_swizzle (group-of-32: and=0x1f, or=0, xor=0x10)
__device__ __forceinline__ float swap16f(float x) {
  return __int_as_float(__builtin_amdgcn_ds_swizzle(__float_as_int(x), 0x401f));
}

__global__ void ohem_init(unsigned* __restrict__ ctrl) {
  for (int i = threadIdx.x; i < C_TOTAL; i += blockDim.x) ctrl[i] = 0u;
}

// Pass 1: per-pixel log-softmax. 16 pixels per wave; channel exp-sum done on the
// matrix pipe via 5 chained V_WMMA_F32_16X16X4_F32 with B = ones (row sums).
__global__ __launch_bounds__(256) void ohem_pixel_pass(
    const float* __restrict__ pred, const int* __restrict__ label,
    float* __restrict__ pbuf, float* __restrict__ nllbuf,
    unsigned* __restrict__ ctrl) {
  __shared__ unsigned hist[1024];
  __shared__ unsigned nvalid;
  for (int i = threadIdx.x; i < 1024; i += 256) hist[i] = 0u;
  if (threadIdx.x == 0) nvalid = 0u;
  __syncthreads();

  const int lane  = threadIdx.x & 31;
  const int half  = lane >> 4;        // f32 A-layout: lanes16-31 hold K=2,3
  const int l     = lane & 15;        // row (pixel) index within tile
  const int gwave = blockIdx.x * 8 + (threadIdx.x >> 5);
  const int nwav  = gridDim.x * 8;
  const int ntile = NPIX / 16;

  const v2f bones = {1.0f, 1.0f};     // B = all-ones 4x16

  for (int t = gwave; t < ntile; t += nwav) {
    const int pbase = t * 16;
    const int p     = pbase + l;
    const int n     = pbase >> HW_SHIFT;
    const int hw    = (pbase & (HWSZ - 1)) + l;
    const size_t base = (size_t)n * ((size_t)CCH * HWSZ) + (size_t)hw;

    // lane (half,l): channels 4j+2*half, 4j+2*half+1 for pixel l, j=0..4 (pad c=19)
    float a[5][2];
    float m = -3.4e38f;
#pragma unroll
    for (int j = 0; j < 5; ++j) {
      const int c0 = 4 * j + 2 * half;
#pragma unroll
      for (int u = 0; u < 2; ++u) {
        const int c = c0 + u;
        float x = (c < CCH) ? pred[base + (size_t)c * HWSZ] : -3.4e38f;
        a[j][u] = x;
        m = fmaxf(m, x);
      }
    }
    m = fmaxf(m, swap16f(m));         // row max across the lane pair

    const int lab = label[p];
    float xl = -3.4e38f;

    v8f acc = {};
#pragma unroll
    for (int j = 0; j < 5; ++j) {
      const int c0 = 4 * j + 2 * half;
      const float e0 = (c0     < CCH) ? __expf(a[j][0] - m) : 0.0f;
      const float e1 = (c0 + 1 < CCH) ? __expf(a[j][1] - m) : 0.0f;
      if (c0     == lab) xl = a[j][0];
      if (c0 + 1 == lab) xl = a[j][1];
      const v2f av = {e0, e1};
      // D[m,n] += sum_k A[m,k] : fp32 row-sum on the matrix pipe
      acc = __builtin_amdgcn_wmma_f32_16x16x4_f32(
          false, av, false, bones, (short)0, acc, false, false);
    }
    xl = fmaxf(xl, swap16f(xl));      // logit of true class (one lane of pair holds it)

    // Extract row-sum for pixel l from C/D layout:
    // lanes0-15: acc[r]=sum(pixel r); lanes16-31: acc[r]=sum(pixel 8+r)
    const int idx = l & 7;
    float s0 = (idx & 1) ? acc[1] : acc[0];
    float s1 = (idx & 1) ? acc[3] : acc[2];
    float s2 = (idx & 1) ? acc[5] : acc[4];
    float s3 = (idx & 1) ? acc[7] : acc[6];
    float u0 = (idx & 2) ? s1 : s0;
    float u1 = (idx & 2) ? s3 : s2;
    float v  = (idx & 4) ? u1 : u0;
    float vx = swap16f(v);
    const float s = (((l >> 3) & 1) == half) ? v : vx;  // sum_c exp(x_c - m), >= 1

    const float logS  = __logf(s);
    const bool  valid = (lab != IGNORE_LBL);
    const float ptrue = __expf(xl - m) / s;
    const float nll   = logS + m - xl;                  // -(x_lab - m - logS)
    const float psort = valid ? ptrue : 2.0f;

    if (half == 0) {
      pbuf[p] = psort;
      if (valid) {
        atomicAdd(&hist[__float_as_uint(psort) >> 20], 1u);  // bits[31:20] <= 0x3F8
        atomicAdd(&nvalid, 1u);
      }
    } else {
      nllbuf[p] = nll;
    }
  }
  __syncthreads();
  for (int i = threadIdx.x; i < 1024; i += 256)
    if (hist[i]) atomicAdd(&ctrl[H1OFF + i], hist[i]);
  if (threadIdx.x == 0 && nvalid) atomicAdd(&ctrl[C_NV], nvalid);
}

// Radix-select scan over a histogram level (single thread; bins <= 4096)
__global__ void ohem_scan(unsigned* __restrict__ ctrl, int stage) {
  if (threadIdx.x != 0 || blockIdx.x != 0) return;
  unsigned k;
  const unsigned* h;
  int nb;
  if (stage == 0) {
    const unsigned nv = ctrl[C_NV];
    const unsigned kk = nv < MIN_KEPT ? nv : MIN_KEPT;
    k = kk ? kk - 1u : 0u;            // k = max(min(MIN_KEPT,nv)-1, 0)
    h = ctrl + H1OFF; nb = 1024;
  } else if (stage == 1) {
    k = ctrl[C_K]; h = ctrl + H2OFF; nb = 4096;
  } else {
    k = ctrl[C_K]; h = ctrl + H3OFF; nb = 256;
  }
  unsigned cum = 0, sel = (unsigned)(nb - 1);
  for (int b = 0; b < nb; ++b) {
    const unsigned c = h[b];
    if (k < cum + c) { sel = (unsigned)b; break; }
    cum += c;
  }
  if (stage == 0)      { ctrl[C_B1] = sel; ctrl[C_K] = k - cum; }
  else if (stage == 1) { ctrl[C_B2] = sel; ctrl[C_K] = k - cum; }
  else {
    const unsigned nv = ctrl[C_NV];
    const unsigned bits = (ctrl[C_B1] << 20) | (ctrl[C_B2] << 8) | sel;
    float th = __uint_as_float(bits);           // exact k-th smallest p_true
    th = th > THRESH ? th : THRESH;
    ctrl[C_TH] = __float_as_uint(th);
    ctrl[C_DO] = nv > MIN_KEPT ? 1u : 0u;
  }
}

// Refine: histogram next radix digit among keys matching the selected prefix.
__global__ __launch_bounds__(256) void ohem_refine(
    const float* __restrict__ pbuf, unsigned* __restrict__ ctrl, int stage) {
  __shared__ unsigned h[4096];
  const int nb = (stage == 0) ? 4096 : 256;
  for (int i = threadIdx.x; i < nb; i += 256) h[i] = 0u;
  __syncthreads();
  const unsigned b1 = ctrl[C_B1], b2 = ctrl[C_B2];
  const unsigned prefix = (stage == 0) ? b1 : ((b1 << 12) | b2);
  const int shift = (stage == 0) ? 20 : 8;
  for (int p = blockIdx.x * 256 + threadIdx.x; p < NPIX; p += gridDim.x * 256) {
    const unsigned bits = __float_as_uint(pbuf[p]);  // invalid (2.0f) never matches
    if ((bits >> shift) == prefix) {
      const unsigned bin = (stage == 0) ? ((bits >> 8) & 0xFFFu) : (bits & 0xFFu);
      atomicAdd(&h[bin], 1u);
    }
  }
  __syncthreads();
  unsigned* gh = (stage == 0) ? (ctrl + H2OFF) : (ctrl + H3OFF);
  for (int i = threadIdx.x; i < nb; i += 256)
    if (h[i]) atomicAdd(&gh[i], h[i]);
}

// Final masked NLL reduction
__global__ __launch_bounds__(256) void ohem_reduce(
    const float* __restrict__ pbuf, const float* __restrict__ nllbuf,
    unsigned* __restrict__ ctrl) {
  __shared__ float    ssum[256];
  __shared__ unsigned scnt[256];
  const float th     = __uint_as_float(ctrl[C_TH]);
  const unsigned doh = ctrl[C_DO];
  float sum = 0.0f; unsigned cnt = 0u;
  for (int p = blockIdx.x * 256 + threadIdx.x; p < NPIX; p += gridDim.x * 256) {
    const float ps = pbuf[p];
    const bool valid = (ps <= 1.0f);                 // invalid stored as 2.0f
    const bool kept  = valid && (doh ? (ps <= th) : true);
    if (kept) { sum += nllbuf[p]; ++cnt; }
  }
  ssum[threadIdx.x] = sum; scnt[threadIdx.x] = cnt;
  __syncthreads();
  for (int s = 128; s > 0; s >>= 1) {
    if (threadIdx.x < s) {
      ssum[threadIdx.x] += ssum[threadIdx.x + s];
      scnt[threadIdx.x] += scnt[threadIdx.x + s];
    }
    __syncthreads();
  }
  if (threadIdx.x == 0) {
    atomicAdd((float*)&ctrl[C_SUM], ssum[0]);
    atomicAdd(&ctrl[C_CNT], scnt[0]);
  }
}

__global__ void ohem_final(const unsigned* __restrict__ ctrl, float* __restrict__ out) {
  const float sum = __uint_as_float(ctrl[C_SUM]);
  const unsigned cnt = ctrl[C_CNT];
  out[0] = sum / (float)(cnt > 0u ? cnt : 1u);
}

extern "C" void kernel_launch(void* const* d_in, const int* in_sizes, int n_in,
                              void* d_out, int out_size, void* d_ws, size_t ws_size,
                              hipStream_t stream) {
  const float* pred  = (const float*)d_in[0];
  const int*   label = (const int*)d_in[1];
  float* out = (float*)d_out;

  float*    pbuf   = (float*)d_ws;                 // 16 MB p_sort keys
  float*    nllbuf = pbuf + NPIX;                  // 16 MB per-pixel NLL
  unsigned* ctrl   = (unsigned*)(nllbuf + NPIX);   // ~21 KB counters

  ohem_init<<<1, 256, 0, stream>>>(ctrl);
  ohem_pixel_pass<<<512, 256, 0, stream>>>(pred, label, pbuf, nllbuf, ctrl);
  ohem_scan<<<1, 1, 0, stream>>>(ctrl, 0);
  ohem_refine<<<1024, 256, 0, stream>>>(pbuf, ctrl, 0);
  ohem_scan<<<1, 1, 0, stream>>>(ctrl, 1);
  ohem_refine<<<1024, 256, 0, stream>>>(pbuf, ctrl, 1);
  ohem_scan<<<1, 1, 0, stream>>>(ctrl, 2);
  ohem_reduce<<<1024, 256, 0, stream>>>(pbuf, nllbuf, ctrl);
  ohem_final<<<1, 1, 0, stream>>>(ctrl, out);
}